// HashEmbedder_43387759624235
// MI455X (gfx1250) — compile-verified
//
#include <hip/hip_runtime.h>
#include <cstdint>

// ---------------------------------------------------------------------------
// Instant-NGP multiresolution hash encoding, forward (MI455X / gfx1250).
//   x:      [N,3]  f32 in [-1,1]
//   tables: [16, 2^19, 2] f32   (64 MB total -> L2-resident on 192 MB L2)
//   out:    [N, 32] f32
//
// Gather-rate bound. Per point: 16 levels x 8 float2 gathers (L2), trilinear
// lerp, then per-wave LDS staging + gfx1250 async LDS->global b128 stores
// with th:TH_STORE_NT so the 128 MB output stream does not evict the tables.
// ---------------------------------------------------------------------------

#define NLV   16
#define TSIZE (1u << 19)
#define TMASK (TSIZE - 1u)
#define PRIME1 2654435761u
#define PRIME2 805459861u

// LDS row stride per point (floats). 36 keeps every 16B chunk 16B-aligned
// (36*4 = 144 = 9*16) and limits ds_store bank conflicts to 2-way.
#define ROWS 36

__device__ __forceinline__ float2 lerp2(float2 a, float2 b, float w, float iw) {
    return make_float2(a.x * iw + b.x * w, a.y * iw + b.y * w);
}

__global__ __launch_bounds__(256) void hash_encode_kernel(
    const float* __restrict__ x,
    const float* __restrict__ tab,
    float* __restrict__ out,
    int n)
{
    // floor(16 * (2^(1/3))^l), l = 0..15
    constexpr float kRes[NLV] = {
        16.f, 20.f, 25.f, 32.f, 40.f, 50.f, 64.f, 80.f,
        101.f, 128.f, 161.f, 203.f, 256.f, 322.f, 406.f, 512.f};

    const int tid  = threadIdx.x;
    const int lane = tid & 31;
    const int wv   = tid >> 5;
    const int gid  = blockIdx.x * 256 + tid;

    // 8 waves * 32 points * 36 floats = 36 KB, wave-private tiles.
    __shared__ float stage[8][32 * ROWS];

    float2 feat[NLV];

    if (gid < n) {
        const float* xp = x + 3u * (size_t)gid;
        const float x0 = __builtin_nontemporal_load(xp + 0);
        const float x1 = __builtin_nontemporal_load(xp + 1);
        const float x2 = __builtin_nontemporal_load(xp + 2);
        // clamped coords (+1 bias pre-applied), and unclamped (+1) for weights
        const float a0 = x0 + 1.0f, a1 = x1 + 1.0f, a2 = x2 + 1.0f;
        const float c0 = fminf(fmaxf(x0, -1.f), 1.f) + 1.0f;
        const float c1 = fminf(fmaxf(x1, -1.f), 1.f) + 1.0f;
        const float c2 = fminf(fmaxf(x2, -1.f), 1.f) + 1.0f;

#pragma unroll
        for (int l = 0; l < NLV; ++l) {
            const float res  = kRes[l];
            const float grid = 2.0f / res;     // compile-time constant
            const float ig   = 1.0f / grid;    // compile-time constant

            const float f0 = floorf(c0 * ig);
            const float f1 = floorf(c1 * ig);
            const float f2 = floorf(c2 * ig);

            // w = (x - vmin)/grid = ((x+1) - f*grid) * ig   (unclamped x)
            const float w0 = (a0 - f0 * grid) * ig;
            const float w1 = (a1 - f1 * grid) * ig;
            const float w2 = (a2 - f2 * grid) * ig;

            // XOR-separable hash; mask the y/z terms once, hx < 2^10 already.
            const uint32_t hx0 = (uint32_t)(int)f0;
            const uint32_t hx1 = hx0 + 1u;
            const uint32_t hyb = (uint32_t)(int)f1 * PRIME1;
            const uint32_t hzb = (uint32_t)(int)f2 * PRIME2;
            const uint32_t hy0 = hyb & TMASK;
            const uint32_t hy1 = (hyb + PRIME1) & TMASK;
            const uint32_t hz0 = hzb & TMASK;
            const uint32_t hz1 = (hzb + PRIME2) & TMASK;
            // share the four x^y combinations; per-corner index = 1 XOR
            const uint32_t a00 = hx0 ^ hy0, a01u = hx0 ^ hy1;
            const uint32_t a10 = hx1 ^ hy0, a11u = hx1 ^ hy1;

            const float2* t = reinterpret_cast<const float2*>(tab) + (size_t)l * TSIZE;

            // 8 corner gathers (global_load_b64, served from L2)
            const float2 e000 = t[a00  ^ hz0];
            const float2 e001 = t[a00  ^ hz1];
            const float2 e010 = t[a01u ^ hz0];
            const float2 e011 = t[a01u ^ hz1];
            const float2 e100 = t[a10  ^ hz0];
            const float2 e101 = t[a10  ^ hz1];
            const float2 e110 = t[a11u ^ hz0];
            const float2 e111 = t[a11u ^ hz1];

            const float iw0 = 1.f - w0, iw1 = 1.f - w1, iw2 = 1.f - w2;

            // trilinear: reduce x, then y, then z (matches reference order)
            const float2 c00v = lerp2(e000, e100, w0, iw0);
            const float2 c01v = lerp2(e001, e101, w0, iw0);
            const float2 c10v = lerp2(e010, e110, w0, iw0);
            const float2 c11v = lerp2(e011, e111, w0, iw0);
            const float2 d0   = lerp2(c00v, c10v, w1, iw1);
            const float2 d1   = lerp2(c01v, c11v, w1, iw1);
            feat[l]           = lerp2(d0, d1, w2, iw2);
        }
    }

    const int wave_first = blockIdx.x * 256 + wv * 32;

    if (wave_first + 32 <= n) {
        // --- fully valid wave: stage in LDS, then async DMA LDS -> global ---
        float* row = &stage[wv][lane * ROWS];
#pragma unroll
        for (int l = 0; l < NLV; ++l) {
            row[2 * l]     = feat[l].x;   // merged into ds_store_b128
            row[2 * l + 1] = feat[l].y;
        }

        // order our ds_stores before the async engine reads LDS
        asm volatile("s_wait_dscnt 0" ::: "memory");

        float* gb = out + (size_t)wave_first * 32;   // 1024 contiguous floats
#pragma unroll
        for (int c = 0; c < 8; ++c) {
            const int g = c * 128 + lane * 4;        // float index in wave tile
            const int p = g >> 5;                    // point within wave
            const int f = g & 31;                    // feature index (mult of 4)
            const uint32_t laddr =
                (uint32_t)(uintptr_t)&stage[wv][p * ROWS + f];   // 16B aligned
            const unsigned long long gaddr =
                (unsigned long long)(uintptr_t)(gb + g);          // 16B aligned
            // gfx1250 async LDS->global store, non-temporal: do not let the
            // 128 MB output stream evict the L2-resident hash tables.
            asm volatile("global_store_async_from_lds_b128 %0, %1, off th:TH_STORE_NT"
                         :: "v"(gaddr), "v"(laddr)
                         : "memory");
        }
        asm volatile("s_wait_asynccnt 0" ::: "memory");
    } else if (gid < n) {
        // tail (never taken for N % 32 == 0): direct guarded stores
        float2* o = reinterpret_cast<float2*>(out + (size_t)gid * 32);
#pragma unroll
        for (int l = 0; l < NLV; ++l) o[l] = feat[l];
    }
}

extern "C" void kernel_launch(void* const* d_in, const int* in_sizes, int n_in,
                              void* d_out, int out_size, void* d_ws, size_t ws_size,
                              hipStream_t stream) {
    const float* x   = (const float*)d_in[0];   // [N,3]
    const float* tab = (const float*)d_in[1];   // [16, 2^19, 2]
    float* out = (float*)d_out;                 // [N,32]

    const int n = in_sizes[0] / 3;
    const int blocks = (n + 255) / 256;
    hipLaunchKernelGGL(hash_encode_kernel, dim3(blocks), dim3(256), 0, stream,
                       x, tab, out, n);
}